// GraphSage_56126632624274
// MI455X (gfx1250) — compile-verified
//
#include <hip/hip_runtime.h>
#include <hip/hip_bf16.h>
#include <stdint.h>

// ---------------------------------------------------------------------------
// GraphSAGE (2-layer, mean aggregation) for MI455X / gfx1250, wave32.
//
// Roofline: GEMMs total ~10 GFLOP (trivial); edge scatter/gather dominates.
// Layer-2 uses the identity  (segsum(h[src])*invdeg) @ Wn == segsum((h@Wn)[src])*invdeg
// to aggregate 64-wide pre-projected features instead of 128-wide -> halves
// layer-2 edge traffic. All dense math on V_WMMA_F32_16X16X4_F32 (exact fp32).
// Atomics resolve in L2 (accumulators fit in the 192MB L2).
// ---------------------------------------------------------------------------

#define NNODES 100000
#define NEDGES 3200000
#define FEAT   128
#define NHID   128
#define NCLASS 64

typedef __attribute__((ext_vector_type(2))) float v2f;
typedef __attribute__((ext_vector_type(8))) float v8f;

// ---------------- utility kernels -----------------------------------------

__global__ __launch_bounds__(256) void zero_f32(float* __restrict__ p, size_t n) {
    size_t i = (size_t)blockIdx.x * blockDim.x + threadIdx.x;
    size_t stride = (size_t)gridDim.x * blockDim.x;
    for (; i < n; i += stride) p[i] = 0.0f;
}

__global__ __launch_bounds__(256) void deg_count(const int* __restrict__ dst,
                                                 float* __restrict__ deg, int ne) {
    int e = blockIdx.x * blockDim.x + threadIdx.x;
    if (e < ne) atomicAdd(&deg[dst[e]], 1.0f);
}

__global__ __launch_bounds__(256) void invert_deg(float* __restrict__ deg, int n) {
    int i = blockIdx.x * blockDim.x + threadIdx.x;
    if (i < n) deg[i] = 1.0f / fmaxf(deg[i], 1.0f);
}

// ---------------- edge scatter-add (dominant cost) -------------------------
// One wave per edge; edge indices are wave-uniform -> readfirstlane so the
// compiler emits scalar loads for src/dst and scalar address math. Gather is
// a coalesced global_load_b128/b64; scatter is global_atomic_add_f32 in L2.

__global__ __launch_bounds__(256) void scatter_add_128(
    const float* __restrict__ feat,  // [NNODES, 128]
    const int*   __restrict__ src,
    const int*   __restrict__ dst,
    float*       __restrict__ agg,   // [NNODES, 128]
    int ne)
{
    long long gid = (long long)blockIdx.x * blockDim.x + threadIdx.x;
    int e = __builtin_amdgcn_readfirstlane((int)(gid >> 5));  // wave-uniform
    if (e >= ne) return;
    int c = ((int)threadIdx.x & 31) << 2;                     // 0..124
    int s = __builtin_amdgcn_readfirstlane(src[e]);
    int d = __builtin_amdgcn_readfirstlane(dst[e]);
    const float4 v = *reinterpret_cast<const float4*>(feat + (size_t)s * 128 + c);
    float* out = agg + (size_t)d * 128 + c;
    atomicAdd(out + 0, v.x);
    atomicAdd(out + 1, v.y);
    atomicAdd(out + 2, v.z);
    atomicAdd(out + 3, v.w);
}

__global__ __launch_bounds__(256) void scatter_add_64(
    const float* __restrict__ feat,  // [NNODES, 64]
    const int*   __restrict__ src,
    const int*   __restrict__ dst,
    float*       __restrict__ agg,   // [NNODES, 64]
    int ne)
{
    long long gid = (long long)blockIdx.x * blockDim.x + threadIdx.x;
    int e = __builtin_amdgcn_readfirstlane((int)(gid >> 5));  // wave-uniform
    if (e >= ne) return;
    int c = ((int)threadIdx.x & 31) << 1;                     // 0..62
    int s = __builtin_amdgcn_readfirstlane(src[e]);
    int d = __builtin_amdgcn_readfirstlane(dst[e]);
    const float2 v = *reinterpret_cast<const float2*>(feat + (size_t)s * 64 + c);
    float* out = agg + (size_t)d * 64 + c;
    atomicAdd(out + 0, v.x);
    atomicAdd(out + 1, v.y);
}

// ---------------- WMMA fragment helpers ------------------------------------
// V_WMMA_F32_16X16X4_F32 layouts (CDNA5 ISA):
//   A 16x4 :  lanes 0-15 -> M=lane, v[0]=K0,v[1]=K1 ; lanes 16-31 -> K2,K3
//   B 4x16 :  lanes 0-15 -> N=lane, v[0]=K0,v[1]=K1 ; lanes 16-31 -> K2,K3
//   C 16x16:  VGPR v: lanes 0-15 -> M=v, N=lane ; lanes 16-31 -> M=v+8

// Layer 1: OUT = relu( X @ Wself + (AGG*invdeg) @ Wneigh + b ), KDIM=NOUT=128
template <int KDIM, int NOUT>
__global__ __launch_bounds__(256) void sage_gemm_dual(
    const float* __restrict__ X, const float* __restrict__ AGG,
    const float* __restrict__ invdeg,
    const float* __restrict__ Wself, const float* __restrict__ Wneigh,
    const float* __restrict__ bias, float* __restrict__ OUT)
{
    constexpr int TN = NOUT / 16;
    const int lane = threadIdx.x & 31;
    const int wave = blockIdx.x * (blockDim.x >> 5) + (threadIdx.x >> 5);
    const int m0 = (wave / TN) * 16;
    const int n0 = (wave % TN) * 16;
    if (m0 >= NNODES) return;

    const int half = lane >> 4;
    const int l15  = lane & 15;
    const int rowA = m0 + l15;
    const int colB = n0 + l15;

    const float idg = invdeg[rowA];
    const float* xrow = X   + (size_t)rowA * KDIM;
    const float* arow = AGG + (size_t)rowA * KDIM;

    v8f c = {};
#pragma unroll 4
    for (int k = 0; k < KDIM; k += 4) {
        const int kk = k + half * 2;
        v2f a_s, a_m, b_s, b_n;
        a_s[0] = xrow[kk];
        a_s[1] = xrow[kk + 1];
        a_m[0] = arow[kk]     * idg;
        a_m[1] = arow[kk + 1] * idg;
        b_s[0] = Wself [(size_t)(kk    ) * NOUT + colB];
        b_s[1] = Wself [(size_t)(kk + 1) * NOUT + colB];
        b_n[0] = Wneigh[(size_t)(kk    ) * NOUT + colB];
        b_n[1] = Wneigh[(size_t)(kk + 1) * NOUT + colB];
        c = __builtin_amdgcn_wmma_f32_16x16x4_f32(false, a_s, false, b_s,
                                                  (short)0, c, false, false);
        c = __builtin_amdgcn_wmma_f32_16x16x4_f32(false, a_m, false, b_n,
                                                  (short)0, c, false, false);
    }

    const float bv = bias[colB];
#pragma unroll
    for (int v = 0; v < 8; ++v) {
        int row = m0 + v + half * 8;
        OUT[(size_t)row * NOUT + colB] = fmaxf(c[v] + bv, 0.0f);
    }
}

// Plain GEMM: OUT = X @ W   (used for z = h @ W2_neigh pre-projection)
template <int KDIM, int NOUT>
__global__ __launch_bounds__(256) void gemm_plain(
    const float* __restrict__ X, const float* __restrict__ W,
    float* __restrict__ OUT)
{
    constexpr int TN = NOUT / 16;
    const int lane = threadIdx.x & 31;
    const int wave = blockIdx.x * (blockDim.x >> 5) + (threadIdx.x >> 5);
    const int m0 = (wave / TN) * 16;
    const int n0 = (wave % TN) * 16;
    if (m0 >= NNODES) return;

    const int half = lane >> 4;
    const int l15  = lane & 15;
    const float* xrow = X + (size_t)(m0 + l15) * KDIM;
    const int colB = n0 + l15;

    v8f c = {};
#pragma unroll 4
    for (int k = 0; k < KDIM; k += 4) {
        const int kk = k + half * 2;
        v2f a, b;
        a[0] = xrow[kk];
        a[1] = xrow[kk + 1];
        b[0] = W[(size_t)(kk    ) * NOUT + colB];
        b[1] = W[(size_t)(kk + 1) * NOUT + colB];
        c = __builtin_amdgcn_wmma_f32_16x16x4_f32(false, a, false, b,
                                                  (short)0, c, false, false);
    }
#pragma unroll
    for (int v = 0; v < 8; ++v) {
        int row = m0 + v + half * 8;
        OUT[(size_t)row * NOUT + colB] = c[v];
    }
}

// Layer-2 finish: OUT = X @ Wself + AGG*invdeg + b
// Accumulator pre-loaded with the already-projected neighbor term (C-fragment
// init), then WMMA accumulates the self path on top. Single pass.
template <int KDIM, int NOUT>
__global__ __launch_bounds__(256) void sage_gemm_final(
    const float* __restrict__ X,       // [NNODES, KDIM]   (h)
    const float* __restrict__ AGG,     // [NNODES, NOUT]   (agg of z)
    const float* __restrict__ invdeg,  // [NNODES]
    const float* __restrict__ Wself,   // [KDIM, NOUT]
    const float* __restrict__ bias,    // [NOUT]
    float*       __restrict__ OUT)     // [NNODES, NOUT]
{
    constexpr int TN = NOUT / 16;
    const int lane = threadIdx.x & 31;
    const int wave = blockIdx.x * (blockDim.x >> 5) + (threadIdx.x >> 5);
    const int m0 = (wave / TN) * 16;
    const int n0 = (wave % TN) * 16;
    if (m0 >= NNODES) return;

    const int half = lane >> 4;
    const int l15  = lane & 15;
    const float* xrow = X + (size_t)(m0 + l15) * KDIM;
    const int colB = n0 + l15;

    v8f c;
#pragma unroll
    for (int v = 0; v < 8; ++v) {
        int row = m0 + v + half * 8;
        c[v] = AGG[(size_t)row * NOUT + colB] * invdeg[row];
    }

#pragma unroll 4
    for (int k = 0; k < KDIM; k += 4) {
        const int kk = k + half * 2;
        v2f a, b;
        a[0] = xrow[kk];
        a[1] = xrow[kk + 1];
        b[0] = Wself[(size_t)(kk    ) * NOUT + colB];
        b[1] = Wself[(size_t)(kk + 1) * NOUT + colB];
        c = __builtin_amdgcn_wmma_f32_16x16x4_f32(false, a, false, b,
                                                  (short)0, c, false, false);
    }

    const float bv = bias[colB];
#pragma unroll
    for (int v = 0; v < 8; ++v) {
        int row = m0 + v + half * 8;
        OUT[(size_t)row * NOUT + colB] = c[v] + bv;
    }
}

// ---------------------------------------------------------------------------

extern "C" void kernel_launch(void* const* d_in, const int* in_sizes, int n_in,
                              void* d_out, int out_size, void* d_ws, size_t ws_size,
                              hipStream_t stream) {
    const float* x        = (const float*)d_in[0];            // [N,128]
    const int*   eidx     = (const int*)  d_in[1];            // [2,E]
    const float* W1_self  = (const float*)d_in[2];            // [128,128]
    const float* W1_neigh = (const float*)d_in[3];
    const float* b1       = (const float*)d_in[4];            // [128]
    const float* W2_self  = (const float*)d_in[5];            // [128,64]
    const float* W2_neigh = (const float*)d_in[6];
    const float* b2       = (const float*)d_in[7];            // [64]
    float*       out      = (float*)d_out;                    // [N,64]

    const int* src = eidx;
    const int* dst = eidx + NEDGES;

    // workspace layout (~103 MB):
    //   invdeg [N] | agg128 [N*128] | h [N*128]
    // after GEMM1, agg128 region is dead -> reuse as z [N*64] + agg64 [N*64]
    float* invdeg = (float*)d_ws;
    size_t off = ((size_t)NNODES + 127) & ~(size_t)127;
    float* agg  = invdeg + off;                               // N*128 floats
    float* h    = agg + (size_t)NNODES * 128;                 // N*128 floats
    float* z    = agg;                                        // N*64 floats
    float* agg2 = agg + (size_t)NNODES * 64;                  // N*64 floats

    const int B = 256;

    // --- degree ---
    zero_f32<<<2048, B, 0, stream>>>(invdeg, (size_t)NNODES);
    deg_count<<<(NEDGES + B - 1) / B, B, 0, stream>>>(dst, invdeg, NEDGES);
    invert_deg<<<(NNODES + B - 1) / B, B, 0, stream>>>(invdeg, NNODES);

    // --- layer 1: aggregate 128-wide, then fused dual-GEMM + relu ---
    zero_f32<<<4096, B, 0, stream>>>(agg, (size_t)NNODES * 128);
    {
        long long nthreads = (long long)NEDGES * 32;
        int blocks = (int)((nthreads + B - 1) / B);
        scatter_add_128<<<blocks, B, 0, stream>>>(x, src, dst, agg, NEDGES);
    }
    {
        int waves = (NNODES / 16) * (NHID / 16);              // 6250 * 8
        sage_gemm_dual<FEAT, NHID><<<waves * 32 / B, B, 0, stream>>>(
            x, agg, invdeg, W1_self, W1_neigh, b1, h);
    }

    // --- layer 2: pre-project, aggregate 64-wide, fused finish ---
    {
        int waves = (NNODES / 16) * (NCLASS / 16);            // 6250 * 4
        gemm_plain<NHID, NCLASS><<<waves * 32 / B, B, 0, stream>>>(h, W2_neigh, z);
    }
    zero_f32<<<4096, B, 0, stream>>>(agg2, (size_t)NNODES * 64);
    {
        long long nthreads = (long long)NEDGES * 32;
        int blocks = (int)((nthreads + B - 1) / B);
        scatter_add_64<<<blocks, B, 0, stream>>>(z, src, dst, agg2, NEDGES);
    }
    {
        int waves = (NNODES / 16) * (NCLASS / 16);            // 6250 * 4
        sage_gemm_final<NHID, NCLASS><<<waves * 32 / B, B, 0, stream>>>(
            h, agg2, invdeg, W2_self, b2, out);
    }
}